// GraycodeEncoder_80358838108496
// MI455X (gfx1250) — compile-verified
//
#include <hip/hip_runtime.h>
#include <hip/hip_bf16.h>

// Graycode encoder: (B,N,3) f32 -> (B,N,96) {0,1} (written as f32).
// Pure store-bandwidth-bound: 192 MB out vs 6 MB in -> ~8.5 us floor @ 23.3 TB/s.
// Strategy: one float4 (4 features) per thread, flat-tid == flat-float4-index so
// every wave issues dense contiguous global_store_b128 with NT hint.

typedef __attribute__((ext_vector_type(4))) float v4f;

#define PTS_PER_BLOCK 16   // block = (24, 16) = 384 threads = 12 waves

__global__ __launch_bounds__(384)
void graycode_encode_kernel(const float* __restrict__ X,
                            float* __restrict__ out,
                            int total_points)
{
    const int j = threadIdx.x;                                  // 0..23 : which float4 of the 96 outputs
    const int p = blockIdx.x * PTS_PER_BLOCK + threadIdx.y;     // point index
    if (p >= total_points) return;

    const int dim = j >> 3;            // 0..2 : coordinate this thread's features come from
    const int k   = (j & 7) << 2;      // 0,4,...,28 : starting feature within the 32-wide group

    // Keep the (tiny, heavily-reused) input stream ahead of the waves.
    if (j == 0) {
        __builtin_prefetch(X + (size_t)p * 3 + 4096, 0, 0);     // speculative gfx1250 global_prefetch_b8
    }

    const float x = X[(size_t)p * 3 + dim];

    // jnp.round == round-half-to-even == v_rndne_f32
    const int   raw  = (int)rintf(x);
    const unsigned v = (unsigned)(raw < 0 ? -raw : raw);
    const unsigned g = v ^ (v >> 1);
    // Prepend sign bit: h bit0 = sign, h bit k = g bit (k-1)  -> feature k = (h>>k)&1, branchless.
    const unsigned h = (g << 1) | (unsigned)(raw > 0);

    v4f o;
    o.x = (float)((h >> (k + 0)) & 1u);
    o.y = (float)((h >> (k + 1)) & 1u);
    o.z = (float)((h >> (k + 2)) & 1u);
    o.w = (float)((h >> (k + 3)) & 1u);

    // Flat float4 index = p*24 + j ; across a block this is exactly contiguous,
    // so each wave stores a dense 512B run. NT: streamed, written once.
    v4f* dst = (v4f*)out + (size_t)p * 24 + j;
    __builtin_nontemporal_store(o, dst);
}

extern "C" void kernel_launch(void* const* d_in, const int* in_sizes, int n_in,
                              void* d_out, int out_size, void* d_ws, size_t ws_size,
                              hipStream_t stream)
{
    const float* X = (const float*)d_in[0];
    float* out     = (float*)d_out;

    const int total_points = in_sizes[0] / 3;            // 8*65536 = 524288
    const int blocks = (total_points + PTS_PER_BLOCK - 1) / PTS_PER_BLOCK;

    dim3 block(24, PTS_PER_BLOCK, 1);
    graycode_encode_kernel<<<blocks, block, 0, stream>>>(X, out, total_points);
}